// SidNetLayer_87883620811425
// MI455X (gfx1250) — compile-verified
//
#include <hip/hip_runtime.h>
#include <hip/hip_bf16.h>
#include <stdint.h>

// SidNet signed diffusion on MI455X (gfx1250, wave32).
// State fits in 192MB L2 -> edge-parallel gather + global_atomic_add_f32 scatter.
// CDNA5 paths: global_load_async_to_lds_b32 staging of edge tiles (ASYNCcnt),
// v_wmma_f32_16x16x4_f32 for the restart-term tile init,
// explicit no-return global_atomic_add_f32 at device scope for the scatter.

#define DFEAT 128
#define TILE 32          // edges per wave tile
#define WAVES_PER_BLK 8  // 256 threads

typedef __attribute__((ext_vector_type(2))) float v2f;
typedef __attribute__((ext_vector_type(8))) float v8f;

// ---------------------------------------------------------------------------
// init: dstP = C*X (via WMMA 16x16 tiles), dstM = 0
// A_r[m,k] = C if m == 4r+k else 0  =>  D = sum_r A_r x B_r = C * X_tile
// Layouts per CDNA5 ISA 7.12.2:
//   A 16x4 f32 : lanes0-15 M=0..15, v0=K0,v1=K1 ; lanes16-31 v0=K2,v1=K3
//   B 4x16 f32 : (mirror of A) lanes0-15 N, v0=K0,v1=K1 ; lanes16-31 v0=K2,v1=K3
//   C/D 16x16  : vgpr v, lanes0-15 -> M=v, N=lane ; lanes16-31 -> M=v+8
// ---------------------------------------------------------------------------
__global__ __launch_bounds__(256)
void sidnet_init_kernel(const float* __restrict__ X,
                        float* __restrict__ dstP, float* __restrict__ dstM,
                        int total, float Cf) {
  const int lane = threadIdx.x & 31;
  const int wave = (blockIdx.x * blockDim.x + threadIdx.x) >> 5;
  const int nWaves = (gridDim.x * blockDim.x) >> 5;
  const int nTiles = total >> 8;           // 256 floats per 16x16 tile
  const int l16 = lane & 15;
  const int hi = lane >> 4;                // 0: lanes 0-15, 1: lanes 16-31

  for (int t = wave; t < nTiles; t += nWaves) {
    const float* chunk = X + (size_t)t * 256;
    v8f acc = {};
#pragma unroll
    for (int r = 0; r < 4; ++r) {
      const int krow = 4 * r + (hi ? 2 : 0);   // K index held in v0 for this lane half
      v2f a;
      a.x = (l16 == krow)     ? Cf : 0.0f;     // A[m=l16, k=krow]
      a.y = (l16 == krow + 1) ? Cf : 0.0f;     // A[m=l16, k=krow+1]
      v2f b;
      b.x = chunk[krow * 16 + l16];            // B[k=krow,   n=l16] = X[4r+k, n]
      b.y = chunk[(krow + 1) * 16 + l16];      // B[k=krow+1, n=l16]
      acc = __builtin_amdgcn_wmma_f32_16x16x4_f32(
          /*neg_a=*/false, a, /*neg_b=*/false, b,
          /*c_mod=*/(short)0, acc, /*reuse_a=*/false, /*reuse_b=*/false);
    }
    float* op = dstP + (size_t)t * 256;
    float* om = dstM + (size_t)t * 256;
#pragma unroll
    for (int v = 0; v < 8; ++v) {
      const int m = v + (hi ? 8 : 0);
      op[m * 16 + l16] = acc[v];
      om[m * 16 + l16] = 0.0f;
    }
  }
  // scalar tail (total not multiple of 256) -- handled by block 0
  if (blockIdx.x == 0) {
    for (int i = (nTiles << 8) + (int)threadIdx.x; i < total; i += 256) {
      dstP[i] = Cf * X[i];
      dstM[i] = 0.0f;
    }
  }
}

// ---------------------------------------------------------------------------
// SpMM scatter: for each edge e of one signed adjacency:
//   dstP[rows[e],:] += vals[e] * gA[cols[e],:]
//   dstM[rows[e],:] += vals[e] * gB[cols[e],:]
// ("+": gA=srcP,gB=srcM ; "-": gA=srcM,gB=srcP)
// Edge triples are double-buffer staged into LDS with async loads.
// Scatter uses explicit no-return global_atomic_add_f32 at device scope.
// ---------------------------------------------------------------------------
__global__ __launch_bounds__(256)
void sidnet_spmm_kernel(const int* __restrict__ rows, const int* __restrict__ cols,
                        const float* __restrict__ vals, int E,
                        const float* __restrict__ gA, const float* __restrict__ gB,
                        float* __restrict__ dstP, float* __restrict__ dstM) {
  // per-wave staging: 2 buffers x (rows|cols|vals) x TILE u32
  __shared__ uint32_t sm[WAVES_PER_BLK][2][3 * TILE];
  const int lane = threadIdx.x & 31;
  const int wIn = threadIdx.x >> 5;
  const int wave = blockIdx.x * WAVES_PER_BLK + wIn;
  const int nWaves = gridDim.x * WAVES_PER_BLK;
  const int nTiles = (E + TILE - 1) / TILE;

  const uint32_t ldsBase = (uint32_t)(uintptr_t)(&sm[wIn][0][0]);

  auto issue = [&](int t, int b) {
    int e = t * TILE + lane;
    if (e >= E) e = E - 1;                        // clamp; tail filtered at use
    const int goff = e * 4;                       // byte offset
    const uint32_t lds = ldsBase + (uint32_t)(b * 3 * TILE * 4) + (uint32_t)(lane * 4);
    asm volatile(
        "global_load_async_to_lds_b32 %0, %1, %2\n\t"
        "global_load_async_to_lds_b32 %3, %1, %4\n\t"
        "global_load_async_to_lds_b32 %5, %1, %6"
        :
        : "v"(lds),                "v"(goff), "s"(rows),
          "v"(lds + TILE * 4),                "s"(cols),
          "v"(lds + 2 * TILE * 4),            "s"(vals)
        : "memory");
  };

  auto process = [&](int t, int b) {
    const uint32_t* s = &sm[wIn][b][0];
    const int cnt = min(TILE, E - t * TILE);
    for (int k = 0; k < cnt; ++k) {
      const uint32_t r = s[k];
      const uint32_t c = s[TILE + k];
      const float v = __uint_as_float(s[2 * TILE + k]);
      const float* pA = gA + (size_t)c * DFEAT + lane;
      const float* pB = gB + (size_t)c * DFEAT + lane;
      const float a0 = pA[0],  a1 = pA[32], a2 = pA[64], a3 = pA[96];
      const float b0 = pB[0],  b1 = pB[32], b2 = pB[64], b3 = pB[96];
      // byte offset of (row r, feature lane) against the SGPR base pointers
      const uint32_t off = r * (DFEAT * 4u) + (uint32_t)lane * 4u;
      asm volatile(
          "global_atomic_add_f32 %0, %1, %5 scope:SCOPE_DEV\n\t"
          "global_atomic_add_f32 %0, %2, %5 offset:128 scope:SCOPE_DEV\n\t"
          "global_atomic_add_f32 %0, %3, %5 offset:256 scope:SCOPE_DEV\n\t"
          "global_atomic_add_f32 %0, %4, %5 offset:384 scope:SCOPE_DEV"
          :
          : "v"(off), "v"(v * a0), "v"(v * a1), "v"(v * a2), "v"(v * a3),
            "s"(dstP)
          : "memory");
      asm volatile(
          "global_atomic_add_f32 %0, %1, %5 scope:SCOPE_DEV\n\t"
          "global_atomic_add_f32 %0, %2, %5 offset:128 scope:SCOPE_DEV\n\t"
          "global_atomic_add_f32 %0, %3, %5 offset:256 scope:SCOPE_DEV\n\t"
          "global_atomic_add_f32 %0, %4, %5 offset:384 scope:SCOPE_DEV"
          :
          : "v"(off), "v"(v * b0), "v"(v * b1), "v"(v * b2), "v"(v * b3),
            "s"(dstM)
          : "memory");
    }
  };

  int t = wave;
  if (t >= nTiles) return;
  issue(t, 0);
  int buf = 0;
  while (t < nTiles) {
    const int tn = t + nWaves;
    if (tn < nTiles) {
      issue(tn, buf ^ 1);
      asm volatile("s_wait_asynccnt 0x3" ::: "memory");  // current tile's 3 done
    } else {
      asm volatile("s_wait_asynccnt 0x0" ::: "memory");
    }
    process(t, buf);
    t = tn;
    buf ^= 1;
  }
}

// ---------------------------------------------------------------------------
extern "C" void kernel_launch(void* const* d_in, const int* in_sizes, int n_in,
                              void* d_out, int out_size, void* d_ws, size_t ws_size,
                              hipStream_t stream) {
  const int*   rows_p = (const int*)  d_in[0];
  const int*   cols_p = (const int*)  d_in[1];
  const float* vals_p = (const float*)d_in[2];
  const int*   rows_m = (const int*)  d_in[3];
  const int*   cols_m = (const int*)  d_in[4];
  const float* vals_m = (const float*)d_in[5];
  const float* X      = (const float*)d_in[6];
  const float* M0     = (const float*)d_in[7];

  const int E  = in_sizes[0];
  const int ND = in_sizes[6];       // N * 128
  const float C = 0.15f;
  const int K = 10;

  float* outP = (float*)d_out;
  float* outM = outP + ND;
  float* wsP  = (float*)d_ws;
  float* wsM  = wsP + ND;

  const float* srcP = X;
  const float* srcM = M0;

  const int initTiles  = ND / 256;
  const int initBlocks = (initTiles + WAVES_PER_BLK - 1) / WAVES_PER_BLK;
  const int spmmBlocks = 640;       // 5120 waves over 18750 edge tiles

  for (int it = 0; it < K; ++it) {
    // even iters -> ws, odd iters -> d_out; K even means iter K-1 lands in d_out
    float* dP = (it & 1) ? outP : wsP;
    float* dM = (it & 1) ? outM : wsM;

    sidnet_init_kernel<<<initBlocks, 256, 0, stream>>>(X, dP, dM, ND, C);
    // "+" adjacency: newP += v*P[c], newM += v*M[c]
    sidnet_spmm_kernel<<<spmmBlocks, 256, 0, stream>>>(rows_p, cols_p, vals_p, E,
                                                       srcP, srcM, dP, dM);
    // "-" adjacency: newP += v*M[c], newM += v*P[c]
    sidnet_spmm_kernel<<<spmmBlocks, 256, 0, stream>>>(rows_m, cols_m, vals_m, E,
                                                       srcM, srcP, dP, dM);
    srcP = dP;
    srcM = dM;
  }
}